// VariableMeanPool_28149215658666
// MI455X (gfx1250) — compile-verified
//
#include <hip/hip_runtime.h>

// ---------------------------------------------------------------------------
// VariableMeanPool on MI455X (gfx1250): memory-bound segmented mean.
// ~130MB of traffic -> ~5.6us at 23.3 TB/s. Pipeline:
//   GLOBAL_LOAD_ASYNC_TO_LDS_B128 (coalesced 512B/wave, ASYNCcnt-tracked)
//   -> ds_load_b128 per-thread contiguous chunks
//   -> register run-accumulation (sorted ids => ~1-2 runs/thread)
//   -> LDS table via ds_add_f32/ds_add_u32 (no-rtn)
//   -> ~65 global_atomic_add_f32/u32 pairs per 4096-element block
//   -> finalize sum / max(count,1).
// WMMA is inapplicable (no matmul structure); the gfx1250-specific paths used
// are the async-tensor copy engine (ASYNCcnt + s_wait_asynccnt), b128 loads,
// native float LDS/global atomics, and wave32 sizing.
// ---------------------------------------------------------------------------

#define BLOCK 256             // 8 wave32 waves per block
#define ITEMS 16              // contiguous elements per thread (64B vals, 64B ids)
#define SPAN  (BLOCK * ITEMS) // 4096 elements per block
#define TABLE 1024            // LDS accumulator slots (span fallback-safe)

__device__ __forceinline__ unsigned lds_offset(const void* p) {
    // Generic pointer to LDS = {aperture_hi32, lds_base_relative_offset_lo32};
    // async-to-LDS VDST wants the LDS-base-relative byte offset.
    return (unsigned)(unsigned long long)(uintptr_t)p;
}

__device__ __forceinline__ void async_load_b128(const void* gaddr, unsigned lds_off) {
    asm volatile("global_load_async_to_lds_b128 %0, %1, off"
                 :
                 : "v"(lds_off), "v"(gaddr)
                 : "memory");
}

__device__ __forceinline__ void wait_asynccnt0() {
    asm volatile("s_wait_asynccnt 0x0" ::: "memory");
}

__global__ void vmp_init_kernel(float* __restrict__ gsum,
                                unsigned* __restrict__ gcnt, int nc) {
    int i = blockIdx.x * blockDim.x + threadIdx.x;
    if (i < nc) { gsum[i] = 0.0f; gcnt[i] = 0u; }
}

__global__ __launch_bounds__(BLOCK)
void vmp_pool_kernel(const float* __restrict__ vals,
                     const int*   __restrict__ ids,
                     float*       __restrict__ gsum,
                     unsigned*    __restrict__ gcnt,
                     int n) {
    __shared__ float    sv[SPAN];      // staged values  (16KB)
    __shared__ int      si[SPAN];      // staged ids     (16KB)
    __shared__ float    ssum[TABLE];   // per-block partial sums (4KB)
    __shared__ unsigned scnt[TABLE];   // per-block partial counts (4KB)
    __shared__ int      sfirst;

    const long long base      = (long long)blockIdx.x * SPAN;
    const bool      full_tile = (base + SPAN) <= (long long)n;   // block-uniform

    // Flush a finished run: LDS table if in-span, direct global atomic otherwise.
    auto flush_global_ok = [&](int first_id, int seg, float a, unsigned k) {
        unsigned d = (unsigned)(seg - first_id);
        if (d < (unsigned)TABLE) {
            __hip_atomic_fetch_add(&ssum[d], a, __ATOMIC_RELAXED,
                                   __HIP_MEMORY_SCOPE_WORKGROUP);
            __hip_atomic_fetch_add(&scnt[d], k, __ATOMIC_RELAXED,
                                   __HIP_MEMORY_SCOPE_WORKGROUP);
        } else {
            __hip_atomic_fetch_add(&gsum[seg], a, __ATOMIC_RELAXED,
                                   __HIP_MEMORY_SCOPE_AGENT);
            __hip_atomic_fetch_add(&gcnt[seg], k, __ATOMIC_RELAXED,
                                   __HIP_MEMORY_SCOPE_AGENT);
        }
    };

    int first_id;

    if (full_tile) {
        // --- Stage the whole tile into LDS with the async copy engine. ------
        // Wave-coalesced: consecutive lanes fetch consecutive 16B chunks, so
        // each instruction issues one contiguous 512B line-aligned request.
        #pragma unroll
        for (int j = 0; j < SPAN / (BLOCK * 4); ++j) {       // 4 iterations
            int e = (j * BLOCK + threadIdx.x) * 4;           // float4 index
            async_load_b128(vals + base + e, lds_offset(&sv[e]));
            async_load_b128(ids  + base + e, lds_offset(&si[e]));
        }

        // Overlap DMA with LDS table zeroing.
        #pragma unroll
        for (int i = threadIdx.x; i < TABLE; i += BLOCK) {
            ssum[i] = 0.0f;
            scnt[i] = 0u;
        }

        wait_asynccnt0();     // my wave's async copies have landed in LDS
        __syncthreads();      // everyone's copies + zeroed table visible

        first_id = si[0];

        // --- Consume: per-thread contiguous 64B from LDS (ds_load_b128). ---
        const int tid0 = threadIdx.x * ITEMS;
        float v[ITEMS];
        int   s[ITEMS];
        const float4* vp = (const float4*)&sv[tid0];
        const int4*   ip = (const int4*)&si[tid0];
        #pragma unroll
        for (int k = 0; k < ITEMS / 4; ++k) {
            float4 f = vp[k];
            int4   q = ip[k];
            v[4*k+0] = f.x; v[4*k+1] = f.y; v[4*k+2] = f.z; v[4*k+3] = f.w;
            s[4*k+0] = q.x; s[4*k+1] = q.y; s[4*k+2] = q.z; s[4*k+3] = q.w;
        }

        int      cur = s[0];
        float    acc = 0.0f;
        unsigned cnt = 0u;
        #pragma unroll
        for (int e = 0; e < ITEMS; ++e) {
            if (s[e] != cur) {
                flush_global_ok(first_id, cur, acc, cnt);
                cur = s[e]; acc = 0.0f; cnt = 0u;
            }
            acc += v[e];
            cnt += 1u;
        }
        flush_global_ok(first_id, cur, acc, cnt);
    } else {
        // --- Partial tail tile (not hit for N = 2^24; kept for generality). -
        #pragma unroll
        for (int i = threadIdx.x; i < TABLE; i += BLOCK) {
            ssum[i] = 0.0f;
            scnt[i] = 0u;
        }
        if (threadIdx.x == 0) {
            long long p = base < (long long)n ? base : (long long)(n - 1);
            sfirst = ids[p];
        }
        __syncthreads();
        first_id = sfirst;

        const long long tbase = base + (long long)threadIdx.x * ITEMS;
        if (tbase < n) {
            const int lim = (int)((n - tbase) < ITEMS ? (n - tbase) : ITEMS);
            int      cur = ids[tbase];
            float    acc = 0.0f;
            unsigned cnt = 0u;
            for (int e = 0; e < lim; ++e) {
                int   se = ids[tbase + e];
                float ve = vals[tbase + e];
                if (se != cur) {
                    flush_global_ok(first_id, cur, acc, cnt);
                    cur = se; acc = 0.0f; cnt = 0u;
                }
                acc += ve;
                cnt += 1u;
            }
            flush_global_ok(first_id, cur, acc, cnt);
        }
    }

    __syncthreads();

    // One global atomic pair per non-empty table slot (~65 per block).
    #pragma unroll
    for (int i = threadIdx.x; i < TABLE; i += BLOCK) {
        unsigned c = scnt[i];
        if (c != 0u) {
            int seg = first_id + i;
            __hip_atomic_fetch_add(&gsum[seg], ssum[i], __ATOMIC_RELAXED,
                                   __HIP_MEMORY_SCOPE_AGENT);
            __hip_atomic_fetch_add(&gcnt[seg], c, __ATOMIC_RELAXED,
                                   __HIP_MEMORY_SCOPE_AGENT);
        }
    }
}

__global__ void vmp_final_kernel(const float* __restrict__ gsum,
                                 const unsigned* __restrict__ gcnt,
                                 float* __restrict__ out, int nc) {
    int i = blockIdx.x * blockDim.x + threadIdx.x;
    if (i < nc) {
        float c = (float)gcnt[i];
        out[i] = gsum[i] / fmaxf(c, 1.0f);
    }
}

extern "C" void kernel_launch(void* const* d_in, const int* in_sizes, int n_in,
                              void* d_out, int out_size, void* d_ws, size_t ws_size,
                              hipStream_t stream) {
    const float* vals = (const float*)d_in[0];
    const int*   ids  = (const int*)d_in[1];
    // d_in[2] is num_crystals; out_size == NC already (host-visible).

    const int n  = in_sizes[0];
    const int nc = out_size;

    float*    gsum = (float*)d_ws;
    unsigned* gcnt = (unsigned*)((char*)d_ws + (size_t)nc * sizeof(float));

    int init_blocks = (nc + BLOCK - 1) / BLOCK;
    vmp_init_kernel<<<init_blocks, BLOCK, 0, stream>>>(gsum, gcnt, nc);

    int pool_blocks = (n + SPAN - 1) / SPAN;   // 4096 for N = 2^24
    vmp_pool_kernel<<<pool_blocks, BLOCK, 0, stream>>>(vals, ids, gsum, gcnt, n);

    int fin_blocks = (nc + BLOCK - 1) / BLOCK;
    vmp_final_kernel<<<fin_blocks, BLOCK, 0, stream>>>(gsum, gcnt, (float*)d_out, nc);
}